// NIMO_2353642078732
// MI455X (gfx1250) — compile-verified
//
#include <hip/hip_runtime.h>

// N=2048 samples, D=128 input/pe dim, H=256 hidden. Output [N, D+1] f32.
#define N_SAMP 2048
#define DDIM   128
#define HDIM   256
#define LN_EPS 1e-5f

typedef float v2f __attribute__((ext_vector_type(2)));
typedef float v8f __attribute__((ext_vector_type(8)));

__device__ __forceinline__ float fast_tanh(float x) {
#if __has_builtin(__builtin_amdgcn_tanhf)
  return __builtin_amdgcn_tanhf(x);
#elif __has_builtin(__builtin_amdgcn_tanh_f32)
  return __builtin_amdgcn_tanh_f32(x);
#else
  float r;
  asm volatile("v_tanh_f32 %0, %1" : "=v"(r) : "v"(x));
  return r;
#endif
}

// sin(z) for |z| <= 0.1 : odd poly, full f32 accuracy (err ~2e-12)
__device__ __forceinline__ float sin_small(float z) {
  float z2 = z * z;
  float p = fmaf(z2, -1.9841270e-4f, 8.3333333e-3f);
  p = fmaf(z2, p, -1.6666667e-1f);
  return fmaf(z * z2, p, z);
}

// ---------------------------------------------------------------------------
// Kernel 1: base[i,h] = sum_k x[i,k] * W1[k,h]   (2048x128 @ 128x256, f32 WMMA)
// One wave32 per 16x16 output tile; K-loop in steps of 4 via
// V_WMMA_F32_16X16X4_F32. VGPR layouts per CDNA5 ISA:
//   A (16x4, 2 VGPRs): elem(m,k): lane = m + 16*(k>>1), vgpr = k&1
//   B (4x16, 2 VGPRs): elem(k,n): lane = n + 16*(k>>1), vgpr = k&1
//   C (16x16, 8 VGPRs): vgpr r: lanes0-15 -> M=r, lanes16-31 -> M=r+8; N=lane&15
// ---------------------------------------------------------------------------
__global__ void __launch_bounds__(32)
base_gemm_wmma(const float* __restrict__ x, const float* __restrict__ W1,
               float* __restrict__ base) {
  const int tile = blockIdx.x;            // 0 .. (128*16 - 1)
  const int tm   = tile / (HDIM / 16);    // M tile 0..127
  const int tn   = tile % (HDIM / 16);    // N tile 0..15
  const int lane = threadIdx.x & 31;
  const int hi   = lane >> 4;             // lane half selects K pair {0,1} vs {2,3}
  const int l16  = lane & 15;
  const int m    = tm * 16 + l16;         // A row for this lane
  const int n    = tn * 16 + l16;         // B/C col for this lane

  v8f acc = {};
#pragma unroll 4
  for (int k0 = 0; k0 < DDIM; k0 += 4) {
    const int k = k0 + 2 * hi;
    v2f a, b;
    a.x = x[m * DDIM + k];
    a.y = x[m * DDIM + k + 1];
    b.x = W1[k * HDIM + n];            // W1 top half: rows 0..127, row-major
    b.y = W1[(k + 1) * HDIM + n];
    acc = __builtin_amdgcn_wmma_f32_16x16x4_f32(
        /*neg_a=*/false, a, /*neg_b=*/false, b,
        /*c_mod=*/(short)0, acc, /*reuse_a=*/false, /*reuse_b=*/false);
  }

  const int row = tm * 16 + 8 * hi;
#pragma unroll
  for (int r = 0; r < 8; ++r)
    base[(row + r) * HDIM + n] = acc[r];
}

// ---------------------------------------------------------------------------
// Kernel 2: g0[j] for j=0..127 (zero-x pass: h1 = W1[D+j,:] + b1).
// One 256-thread block per j; LDS tree reductions for LN stats and W2 dot.
// ---------------------------------------------------------------------------
__device__ __forceinline__ float block_reduce256(float v, float* buf) {
  const int t = threadIdx.x;
  buf[t] = v;
  __syncthreads();
#pragma unroll
  for (int s = 128; s > 0; s >>= 1) {
    if (t < s) buf[t] += buf[t + s];
    __syncthreads();
  }
  float r = buf[0];
  __syncthreads();
  return r;
}

__global__ void __launch_bounds__(HDIM)
g0_kernel(const float* __restrict__ W1, const float* __restrict__ b1,
          const float* __restrict__ ln_g, const float* __restrict__ ln_b,
          const float* __restrict__ W2, const float* __restrict__ b2,
          float* __restrict__ g0) {
  __shared__ float buf[HDIM];
  const int j = blockIdx.x;
  const int h = threadIdx.x;

  const float h1 = W1[(DDIM + j) * HDIM + h] + b1[h];
  const float a  = sin_small(0.1f * fast_tanh(h1));

  const float s1  = block_reduce256(a, buf);
  const float s2  = block_reduce256(a * a, buf);
  const float mu  = s1 * (1.0f / HDIM);
  const float var = fmaf(s2, 1.0f / HDIM, -mu * mu);
  const float rs  = rsqrtf(var + LN_EPS);

  const float ln  = fmaf((a - mu) * rs, ln_g[h], ln_b[h]);
  const float dot = block_reduce256(ln * W2[h], buf);
  if (h == 0) g0[j] = dot + b2[0];
}

// ---------------------------------------------------------------------------
// Kernel 3: fused rank-1 correction + tanh/sin + LayerNorm + W2 dot + output.
// One wave32 per (i,j) row; 8 H-elements per lane (stride-32, coalesced);
// wave shuffle reductions for LN stats and the W2 dot.
// ---------------------------------------------------------------------------
__global__ void __launch_bounds__(256)
fused_rows(const float* __restrict__ x, const float* __restrict__ W1,
           const float* __restrict__ b1, const float* __restrict__ ln_g,
           const float* __restrict__ ln_b, const float* __restrict__ W2,
           const float* __restrict__ b2, const float* __restrict__ base,
           const float* __restrict__ g0, float* __restrict__ out) {
  const int wave = threadIdx.x >> 5;               // 0..7
  const int lane = threadIdx.x & 31;
  const int row  = blockIdx.x * 8 + wave;          // 0 .. N*D-1
  const int i    = row >> 7;                       // row / 128
  const int j    = row & (DDIM - 1);               // row % 128

  const float xs = x[i * DDIM + j];
  const float* __restrict__ brow = base + i * HDIM;
  const float* __restrict__ w1j  = W1 + j * HDIM;           // -x_ij * this row
  const float* __restrict__ w1p  = W1 + (DDIM + j) * HDIM;  // + PE row

  float a[8];
  float s1 = 0.0f, s2 = 0.0f;
#pragma unroll
  for (int t = 0; t < 8; ++t) {
    const int h = lane + 32 * t;
    float h1 = brow[h] - xs * w1j[h] + w1p[h] + b1[h];
    float av = sin_small(0.1f * fast_tanh(h1));
    a[t] = av;
    s1 += av;
    s2 = fmaf(av, av, s2);
  }
#pragma unroll
  for (int msk = 16; msk >= 1; msk >>= 1) {
    s1 += __shfl_xor(s1, msk, 32);
    s2 += __shfl_xor(s2, msk, 32);
  }
  const float mu  = s1 * (1.0f / HDIM);
  const float var = fmaf(s2, 1.0f / HDIM, -mu * mu);
  const float rs  = rsqrtf(var + LN_EPS);

  float dot = 0.0f;
#pragma unroll
  for (int t = 0; t < 8; ++t) {
    const int h = lane + 32 * t;
    const float ln = fmaf((a[t] - mu) * rs, ln_g[h], ln_b[h]);
    dot = fmaf(ln, W2[h], dot);
  }
#pragma unroll
  for (int msk = 16; msk >= 1; msk >>= 1) dot += __shfl_xor(dot, msk, 32);

  if (lane == 0) {
    const float g = dot + b2[0];
    out[i * (DDIM + 1) + 1 + j] = xs * (1.0f + g - g0[j]);
    if (j == 0) out[i * (DDIM + 1)] = 1.0f;   // B0 column
  }
}

// ---------------------------------------------------------------------------
extern "C" void kernel_launch(void* const* d_in, const int* in_sizes, int n_in,
                              void* d_out, int out_size, void* d_ws, size_t ws_size,
                              hipStream_t stream) {
  const float* x    = (const float*)d_in[0];  // [2048,128]
  const float* W1   = (const float*)d_in[1];  // [256,256]
  const float* b1   = (const float*)d_in[2];  // [256]
  const float* ln_g = (const float*)d_in[3];  // [256]
  const float* ln_b = (const float*)d_in[4];  // [256]
  const float* W2   = (const float*)d_in[5];  // [256,1]
  const float* b2   = (const float*)d_in[6];  // [1]
  float* out = (float*)d_out;                 // [2048,129]

  float* base = (float*)d_ws;                 // [2048,256] = 2 MB
  float* g0   = base + N_SAMP * HDIM;         // [128]

  // base = x @ W1[0:128,:]  (f32 WMMA, one wave per 16x16 tile)
  base_gemm_wmma<<<(N_SAMP / 16) * (HDIM / 16), 32, 0, stream>>>(x, W1, base);
  // g0[j] (independent of base; ordered on the same stream anyway)
  g0_kernel<<<DDIM, HDIM, 0, stream>>>(W1, b1, ln_g, ln_b, W2, b2, g0);
  // fused main pass: 262144 rows, 8 waves/block
  fused_rows<<<(N_SAMP * DDIM) / 8, 256, 0, stream>>>(
      x, W1, b1, ln_g, ln_b, W2, b2, base, g0, out);
}